// GNN_10213432230582
// MI455X (gfx1250) — compile-verified
//
#include <hip/hip_runtime.h>
#include <hip/hip_bf16.h>

// GCN (8 GCNConv layers) for MI455X / gfx1250, fp32 throughout.
// GEMMs run on the tensor pipe via V_WMMA_F32_16X16X4_F32 (wave32 WMMA),
// with a register-software-pipelined fragment stream so WMMAs overlap
// outstanding global loads instead of draining loadcnt each step.

typedef float v2f __attribute__((ext_vector_type(2)));
typedef float v8f __attribute__((ext_vector_type(8)));

#define HID   128
#define MTILE 4      // 4 x 16 = 64 rows per block; B fragment reused 4x
#define GRP   4      // K-steps per pipeline stage (GRP*4 = 16 K-values)
#define NGRP  (HID / (4 * GRP))   // 8 pipeline stages

// ---------------- normalization precompute ----------------

__global__ void deg_init_kernel(float* __restrict__ deg, int n) {
    int i = blockIdx.x * blockDim.x + threadIdx.x;
    if (i < n) deg[i] = 1.0f;  // self-loop weight
}

__global__ void deg_edge_kernel(float* __restrict__ deg,
                                const long long* __restrict__ dst,
                                const float* __restrict__ ew, int e) {
    int i = blockIdx.x * blockDim.x + threadIdx.x;
    if (i < e) atomicAdd(&deg[(int)dst[i]], ew[i]);
}

// deg -> dis (in place), selfnorm = dis*dis
__global__ void dis_kernel(float* __restrict__ deg,
                           float* __restrict__ selfnorm, int n) {
    int i = blockIdx.x * blockDim.x + threadIdx.x;
    if (i < n) {
        float d  = deg[i];
        float di = (d > 0.0f) ? rsqrtf(d) : 0.0f;
        deg[i] = di;
        selfnorm[i] = di * di;
    }
}

__global__ void norm_edge_kernel(const float* __restrict__ dis,
                                 const float* __restrict__ ew,
                                 const long long* __restrict__ src,
                                 const long long* __restrict__ dst,
                                 float* __restrict__ norm, int e) {
    int i = blockIdx.x * blockDim.x + threadIdx.x;
    if (i < e) norm[i] = dis[(int)src[i]] * ew[i] * dis[(int)dst[i]];
}

// ---------------- WMMA fp32 GEMM: O[N,128] = act(A[N,128]) @ W[128,128] ----
// Block = 256 threads = 8 waves; block owns 64 output rows (4 M-tiles),
// each wave a 16-wide column tile. Double-buffered A/B fragment pipeline:
// loads for stage g+1 issue before the 16 WMMAs of stage g.
// A-row pointers are clamped (EXEC must stay all-ones for WMMA); only the
// stores are row-guarded.

template <bool RELU>
__global__ __launch_bounds__(256) void gemm128_wmma_kernel(
        const float* __restrict__ A, const float* __restrict__ W,
        float* __restrict__ O, int n) {
    const int lane    = threadIdx.x & 31;
    const int wave    = threadIdx.x >> 5;        // 0..7 -> column tile
    const int colbase = wave * 16;
    const int m  = lane & 15;                    // row (A) / col (B) in tile
    const int kh = (lane >> 4) * 2;              // lane-half K offset (0 or 2)
    const int rowblock = blockIdx.x * (16 * MTILE);

    const float* __restrict__ arow[MTILE];
    #pragma unroll
    for (int t = 0; t < MTILE; ++t) {
        int r = rowblock + t * 16 + m;
        if (r > n - 1) r = n - 1;                // clamp, keep EXEC full
        arow[t] = A + (size_t)r * HID;
    }
    // per-lane B base: column colbase+m, row kh
    const float* __restrict__ bbase = W + (size_t)kh * HID + colbase + m;

    v8f acc[MTILE];
    #pragma unroll
    for (int t = 0; t < MTILE; ++t) acc[t] = (v8f){};

    v2f a0[GRP][MTILE], a1[GRP][MTILE];
    v2f b0[GRP], b1[GRP];

    // stage g covers K-steps g*GRP .. g*GRP+GRP-1 (K-step s => K base s*4)
    auto load_stage = [&](v2f (&ab)[GRP][MTILE], v2f (&bb)[GRP], int g) {
        #pragma unroll
        for (int j = 0; j < GRP; ++j) {
            const int ks = g * GRP + j;          // K-step index
            bb[j].x = bbase[(size_t)(ks * 4) * HID];
            bb[j].y = bbase[(size_t)(ks * 4 + 1) * HID];
            #pragma unroll
            for (int t = 0; t < MTILE; ++t)
                ab[j][t] = *(const v2f*)(arow[t] + ks * 4 + kh);
        }
    };
    auto wmma_stage = [&](v2f (&ab)[GRP][MTILE], v2f (&bb)[GRP]) {
        #pragma unroll
        for (int j = 0; j < GRP; ++j) {
            #pragma unroll
            for (int t = 0; t < MTILE; ++t) {
                v2f a = ab[j][t];
                if (RELU) {
                    a.x = fmaxf(a.x, 0.0f);
                    a.y = fmaxf(a.y, 0.0f);
                }
                // (neg_a, A, neg_b, B, c_mod, C, reuse_a, reuse_b)
                acc[t] = __builtin_amdgcn_wmma_f32_16x16x4_f32(
                             false, a, false, bb[j], (short)0, acc[t],
                             false, false);
            }
        }
    };

    load_stage(a0, b0, 0);
    #pragma unroll
    for (int g = 0; g < NGRP; g += 2) {
        load_stage(a1, b1, g + 1);               // prefetch next stage
        wmma_stage(a0, b0);                      // compute current stage
        if (g + 2 < NGRP) load_stage(a0, b0, g + 2);
        wmma_stage(a1, b1);
    }

    // C/D layout: VGPR v -> row M = v + 8*(lane/16), col N = lane%16
    const int c = colbase + m;
    #pragma unroll
    for (int t = 0; t < MTILE; ++t) {
        const int r0 = rowblock + t * 16 + (lane >> 4) * 8;
        #pragma unroll
        for (int v = 0; v < 8; ++v) {
            int r = r0 + v;
            if (r < n) O[(size_t)r * HID + c] = acc[t][v];
        }
    }
}

// ---------------- aggregation ----------------

// AGG[i,f] = selfnorm[i]*HW[i,f] + b[f]   (doubles as zero-init for atomics)
__global__ void agg_init_kernel(const float* __restrict__ HW,
                                const float* __restrict__ selfnorm,
                                const float* __restrict__ b,
                                float* __restrict__ AGG, int n) {
    int i = blockIdx.x * blockDim.x + threadIdx.x;
    if (i < n * HID) {
        int node = i >> 7, f = i & (HID - 1);
        AGG[i] = selfnorm[node] * HW[i] + b[f];
    }
}

// One wave per edge: float4 gather from HW[src], 4 atomic f32 adds to AGG[dst].
__global__ __launch_bounds__(256) void agg_edge_kernel(
        const float* __restrict__ HW, const float* __restrict__ norm,
        const long long* __restrict__ src, const long long* __restrict__ dst,
        float* __restrict__ AGG, int e) {
    int edge = blockIdx.x * 8 + (threadIdx.x >> 5);
    if (edge >= e) return;
    int lane = threadIdx.x & 31;
    float nv = norm[edge];
    int s = (int)src[edge], d = (int)dst[edge];
    float4 h = ((const float4*)(HW + (size_t)s * HID))[lane];
    float* o = AGG + (size_t)d * HID + lane * 4;
    atomicAdd(o + 0, nv * h.x);
    atomicAdd(o + 1, nv * h.y);
    atomicAdd(o + 2, nv * h.z);
    atomicAdd(o + 3, nv * h.w);
}

// ---------------- layer 8 (D_OUT = 2, too narrow for WMMA) ----------------

__global__ void gemm_out_kernel(const float* __restrict__ H,
                                const float* __restrict__ W8,
                                float* __restrict__ HWo, int n) {
    int i = blockIdx.x * blockDim.x + threadIdx.x;
    if (i >= n) return;
    const float* hr = H + (size_t)i * HID;
    float a0 = 0.0f, a1 = 0.0f;
    #pragma unroll 4
    for (int k = 0; k < HID; ++k) {
        float h = fmaxf(hr[k], 0.0f);   // ReLU of layer-7 output
        a0 = fmaf(h, W8[k * 2 + 0], a0);
        a1 = fmaf(h, W8[k * 2 + 1], a1);
    }
    HWo[i * 2 + 0] = a0;
    HWo[i * 2 + 1] = a1;
}

__global__ void out_init_kernel(const float* __restrict__ HWo,
                                const float* __restrict__ selfnorm,
                                const float* __restrict__ b8,
                                float* __restrict__ out, int n) {
    int i = blockIdx.x * blockDim.x + threadIdx.x;
    if (i < n * 2) {
        int node = i >> 1, f = i & 1;
        out[i] = selfnorm[node] * HWo[i] + b8[f];
    }
}

__global__ void out_edge_kernel(const float* __restrict__ HWo,
                                const float* __restrict__ norm,
                                const long long* __restrict__ src,
                                const long long* __restrict__ dst,
                                float* __restrict__ out, int e) {
    int i = blockIdx.x * blockDim.x + threadIdx.x;
    if (i >= e) return;
    float nv = norm[i];
    int s = (int)src[i], d = (int)dst[i];
    atomicAdd(&out[d * 2 + 0], nv * HWo[s * 2 + 0]);
    atomicAdd(&out[d * 2 + 1], nv * HWo[s * 2 + 1]);
}

// ---------------- host orchestration ----------------

extern "C" void kernel_launch(void* const* d_in, const int* in_sizes, int n_in,
                              void* d_out, int out_size, void* d_ws, size_t ws_size,
                              hipStream_t stream) {
    const float*     x    = (const float*)d_in[0];
    const long long* ei   = (const long long*)d_in[1];   // int64 (2,E)
    const float*     ea   = (const float*)d_in[2];
    const float*     W1   = (const float*)d_in[3];
    const float*     b1   = (const float*)d_in[4];
    const float*     Wmid = (const float*)d_in[5];       // (6,128,128)
    const float*     bmid = (const float*)d_in[6];       // (6,128)
    const float*     W8   = (const float*)d_in[7];       // (128,2)
    const float*     b8   = (const float*)d_in[8];
    float* out = (float*)d_out;

    const int N = in_sizes[0] / HID;
    const int E = in_sizes[1] / 2;
    const long long* src = ei;
    const long long* dst = ei + E;

    // workspace layout (floats)
    float* f        = (float*)d_ws;
    float* bufA     = f;                        // N*128 activations
    float* bufB     = bufA + (size_t)N * HID;   // N*128 H@W
    float* normw    = bufB + (size_t)N * HID;   // E
    float* deg      = normw + E;                // N  (becomes dis)
    float* selfnorm = deg + N;                  // N
    float* HWo      = selfnorm + N;             // N*2

    const int T = 256;
    auto cdiv = [](int a, int b) { return (a + b - 1) / b; };
    const int gemm_blocks = cdiv(N, 16 * MTILE);

    // ---- normalization (layer-invariant) ----
    deg_init_kernel<<<cdiv(N, T), T, 0, stream>>>(deg, N);
    deg_edge_kernel<<<cdiv(E, T), T, 0, stream>>>(deg, dst, ea, E);
    dis_kernel<<<cdiv(N, T), T, 0, stream>>>(deg, selfnorm, N);
    norm_edge_kernel<<<cdiv(E, T), T, 0, stream>>>(deg, ea, src, dst, normw, E);

    // ---- layer 1: input x, no input ReLU ----
    gemm128_wmma_kernel<false><<<gemm_blocks, T, 0, stream>>>(x, W1, bufB, N);
    agg_init_kernel<<<cdiv(N * HID, T), T, 0, stream>>>(bufB, selfnorm, b1, bufA, N);
    agg_edge_kernel<<<cdiv(E, 8), T, 0, stream>>>(bufB, normw, src, dst, bufA, E);

    // ---- layers 2..7: ReLU folded into GEMM A-load ----
    for (int l = 0; l < 6; ++l) {
        gemm128_wmma_kernel<true><<<gemm_blocks, T, 0, stream>>>(
            bufA, Wmid + (size_t)l * HID * HID, bufB, N);
        agg_init_kernel<<<cdiv(N * HID, T), T, 0, stream>>>(
            bufB, selfnorm, bmid + (size_t)l * HID, bufA, N);
        agg_edge_kernel<<<cdiv(E, 8), T, 0, stream>>>(bufB, normw, src, dst, bufA, E);
    }

    // ---- layer 8: D_OUT=2, no output ReLU ----
    gemm_out_kernel<<<cdiv(N, T), T, 0, stream>>>(bufA, W8, HWo, N);
    out_init_kernel<<<cdiv(N * 2, T), T, 0, stream>>>(HWo, selfnorm, b8, out, N);
    out_edge_kernel<<<cdiv(E, T), T, 0, stream>>>(HWo, normw, src, dst, out, E);
}